// Gated_49941879717984
// MI455X (gfx1250) — compile-verified
//
#include <hip/hip_runtime.h>

typedef __attribute__((ext_vector_type(2))) float v2f;
typedef __attribute__((ext_vector_type(8))) float v8f;

#define DCOLS 512
#define WAVES_PER_BLOCK 8
#define CHUNK 1024

// ---------------------------------------------------------------------------
// Kernel 1: scores[i] = dot(V[i,:], W) + b   via V_WMMA_F32_16X16X4_F32.
// One wave32 owns a 16-row tile of V. A-fragment layout (ISA 7.12.2, 32-bit
// A 16x4): lanes 0-15 hold K={k,k+1} for M=lane, lanes 16-31 hold K={k+2,k+3}.
// B is the W chunk broadcast across all 16 N-columns with the matching K
// striping, so D[m][n] == dot(V_row_m, W) in every column n.
// ---------------------------------------------------------------------------
__global__ __launch_bounds__(256) void gated_scores_wmma(
    const float* __restrict__ V, const float* __restrict__ W,
    const float* __restrict__ bias, float* __restrict__ scores, int n_nodes)
{
    const int lane    = threadIdx.x & 31;
    const int wave    = threadIdx.x >> 5;
    const int tile    = blockIdx.x * WAVES_PER_BLOCK + wave;
    const int row_base = tile * 16;
    const int lane_lo = lane & 15;
    const int sel     = lane >> 4;          // 0: lanes 0-15, 1: lanes 16-31

    int row = row_base + lane_lo;
    if (row >= n_nodes) row = n_nodes - 1;  // clamp loads; stores are masked
    const float* arow = V + (size_t)row * DCOLS + 2 * sel;

    v8f c = {};
    #pragma unroll 4
    for (int k = 0; k < DCOLS; k += 4) {
        v2f a = *(const v2f*)(arow + k);            // A[m][k+2sel .. k+2sel+1]
        v2f b = *(const v2f*)(W + k + 2 * sel);     // B rows broadcast over N
        c = __builtin_amdgcn_wmma_f32_16x16x4_f32(
                /*neg_a=*/false, a, /*neg_b=*/false, b,
                /*c_mod=*/(short)0, c, /*reuse_a=*/false, /*reuse_b=*/false);
    }

    // C/D layout: VGPR r, lanes 0-15 -> M=r (N=lane), lanes 16-31 -> M=r+8.
    // Every column is identical; lanes with N==0 (lanes 0 and 16) write out.
    const float bv = bias[0];
    if (lane_lo == 0) {
        #pragma unroll
        for (int r = 0; r < 8; ++r) {
            int m = row_base + sel * 8 + r;
            if (m < n_nodes) scores[m] = c[r] + bv;
        }
    }
}

// ---------------------------------------------------------------------------
// Kernel 2: one block per graph. batch_index is sorted, so the segment is a
// contiguous range found by binary search. Block-reduce max and sum(exp),
// then each thread owns 2 of 512 output columns and streams the segment's
// V rows with coalesced float2 loads, accumulating alpha_i * V_i.
// ---------------------------------------------------------------------------
__global__ __launch_bounds__(256) void gated_softmax_pool(
    const float* __restrict__ V, const float* __restrict__ scores,
    const int* __restrict__ bidx, float* __restrict__ H, int n_nodes)
{
    __shared__ float red[256];
    __shared__ float alpha[CHUNK];
    const int g   = blockIdx.x;
    const int tid = threadIdx.x;

    // lower_bound(g) and lower_bound(g+1) over sorted batch_index
    int lo = 0, hi = n_nodes;
    while (lo < hi) { int mid = (lo + hi) >> 1; if (bidx[mid] <  g)     lo = mid + 1; else hi = mid; }
    const int s = lo;
    hi = n_nodes;
    while (lo < hi) { int mid = (lo + hi) >> 1; if (bidx[mid] <  g + 1) lo = mid + 1; else hi = mid; }
    const int e = lo;
    const int n = e - s;

    float* out = H + (size_t)g * DCOLS;
    if (n == 0) {                       // empty segment -> zeros
        out[2 * tid]     = 0.0f;
        out[2 * tid + 1] = 0.0f;
        return;
    }

    // ---- segment max ----
    float m = -INFINITY;
    for (int i = s + tid; i < e; i += 256) m = fmaxf(m, scores[i]);
    red[tid] = m; __syncthreads();
    for (int off = 128; off > 0; off >>= 1) {
        if (tid < off) red[tid] = fmaxf(red[tid], red[tid + off]);
        __syncthreads();
    }
    m = red[0]; __syncthreads();

    // ---- segment sum(exp) ----
    float ssum = 0.0f;
    for (int i = s + tid; i < e; i += 256) ssum += __expf(scores[i] - m);
    red[tid] = ssum; __syncthreads();
    for (int off = 128; off > 0; off >>= 1) {
        if (tid < off) red[tid] += red[tid + off];
        __syncthreads();
    }
    const float inv_denom = 1.0f / red[0];
    __syncthreads();

    // ---- weighted accumulation: each thread owns columns 2*tid, 2*tid+1 ----
    float accx = 0.0f, accy = 0.0f;
    for (int c0 = 0; c0 < n; c0 += CHUNK) {
        const int cn = min(CHUNK, n - c0);
        for (int i = tid; i < cn; i += 256)
            alpha[i] = __expf(scores[s + c0 + i] - m) * inv_denom;
        __syncthreads();

        const float* rowp = V + (size_t)(s + c0) * DCOLS + 2 * tid;
        for (int i = 0; i < cn; ++i) {
            if (i + 2 < cn)   // global_prefetch_b8 two rows ahead
                __builtin_prefetch(rowp + (size_t)(i + 2) * DCOLS, 0, 0);
            const float  a = alpha[i];
            const float2 v = *(const float2*)(rowp + (size_t)i * DCOLS);
            accx = fmaf(a, v.x, accx);
            accy = fmaf(a, v.y, accy);
        }
        __syncthreads();
    }
    out[2 * tid]     = accx;
    out[2 * tid + 1] = accy;
}

extern "C" void kernel_launch(void* const* d_in, const int* in_sizes, int n_in,
                              void* d_out, int out_size, void* d_ws, size_t ws_size,
                              hipStream_t stream) {
    const float* V    = (const float*)d_in[0];   // [N, 512]
    const float* W    = (const float*)d_in[1];   // [1, 512]
    const float* bl   = (const float*)d_in[2];   // [1]
    const int*   bidx = (const int*)  d_in[3];   // [N], sorted
    const int n_nodes    = in_sizes[3];
    const int num_graphs = out_size / DCOLS;     // device scalar unreadable under capture

    float* scores = (float*)d_ws;                // N floats (~800 KB) of scratch

    const int tiles  = (n_nodes + 15) / 16;
    const int blocks = (tiles + WAVES_PER_BLOCK - 1) / WAVES_PER_BLOCK;
    gated_scores_wmma<<<blocks, 256, 0, stream>>>(V, W, bl, scores, n_nodes);
    gated_softmax_pool<<<num_graphs, 256, 0, stream>>>(V, scores, bidx,
                                                       (float*)d_out, n_nodes);
}